// nUIV_NODE_68410239090966
// MI455X (gfx1250) — compile-verified
//
#include <hip/hip_runtime.h>
#include <hip/hip_bf16.h>
#include <stddef.h>

// CDNA5 / gfx1250. wave32 only.
typedef __attribute__((ext_vector_type(2))) float v2f;
typedef __attribute__((ext_vector_type(8))) float v8f;

// ---------------------------------------------------------------------------
// prep_matrix: TS = ts*ts, row sums -> norm, diag.  One block per row.
// ---------------------------------------------------------------------------
__global__ void prep_matrix(const float* __restrict__ ts,
                            float* __restrict__ TS,
                            float* __restrict__ nrm,
                            float* __restrict__ dg,
                            int N) {
    __shared__ float red[256];
    const int row = blockIdx.x;
    const size_t base = (size_t)row * N;
    float s = 0.f;
    for (int j = threadIdx.x; j < N; j += blockDim.x) {
        float v = ts[base + j];
        float q = v * v;
        TS[base + j] = q;
        s += q;
    }
    red[threadIdx.x] = s;
    __syncthreads();
    for (int off = 128; off > 0; off >>= 1) {
        if (threadIdx.x < off) red[threadIdx.x] += red[threadIdx.x + off];
        __syncthreads();
    }
    if (threadIdx.x == 0) {
        float tv = ts[base + row];
        float d = tv * tv;          // recompute diag locally (no global visibility dependence)
        dg[row] = d;
        float nm = d / (red[0] - d);
        nrm[row] = nm < 1.f ? nm : 1.f;
    }
}

// ---------------------------------------------------------------------------
// prep_state: s0 = x0^2 (interleaved (N,3) in x0 -> SoA), nv0, traj[0].
// ---------------------------------------------------------------------------
__global__ void prep_state(const float* __restrict__ x0,
                           const float* __restrict__ nrm,
                           float* __restrict__ U,  float* __restrict__ I,  float* __restrict__ V,
                           float* __restrict__ Us, float* __restrict__ Is, float* __restrict__ Vs,
                           float* __restrict__ nv, float* __restrict__ traj, int N) {
    int i = blockIdx.x * blockDim.x + threadIdx.x;
    if (i >= N) return;
    float u = x0[3 * i + 0]; u *= u;
    float w = x0[3 * i + 1]; w *= w;
    float v = x0[3 * i + 2]; v *= v;
    U[i] = u;  I[i] = w;  V[i] = v;
    Us[i] = u; Is[i] = w; Vs[i] = v;
    nv[i] = nrm[i] * v;
    traj[(size_t)0 * N + i] = u;
    traj[(size_t)1 * N + i] = w;
    traj[(size_t)2 * N + i] = v;
}

// ---------------------------------------------------------------------------
// stage_kernel: one RK4 stage.
//   coupling = nv_in @ TS   via V_WMMA_F32_16X16X4_F32 (exact fp32 MACs)
//   then per-node k, RK4 accumulation, next-stage state + nv_out.
// Block = 256 threads = 8 waves.  Each block owns 2 column slabs of 16;
// 4 waves split the K (row) range per slab, partials reduced through LDS.
// Templated on N so the unrolled B-tile loads use immediate IOFFSETs
// (stride 4*N floats = 64KB for N=4096) instead of per-iter 64-bit adds.
// ---------------------------------------------------------------------------
template <int NC>
__global__ void stage_kernel(const float* __restrict__ TS,
                             const float* __restrict__ nrm,
                             const float* __restrict__ dg,
                             const float* __restrict__ betas,
                             const float* __restrict__ deltas,
                             const float* __restrict__ cs,
                             const float* __restrict__ ps,
                             float* __restrict__ U,  float* __restrict__ I,  float* __restrict__ V,
                             float* __restrict__ Us, float* __restrict__ Is, float* __restrict__ Vs,
                             const float* __restrict__ nv_in,
                             float* __restrict__ nv_out,
                             float* __restrict__ ksU, float* __restrict__ ksI, float* __restrict__ ksV,
                             float* __restrict__ traj,
                             const float* __restrict__ times,
                             int t, int stage, int Nrt) {
    const int N = (NC > 0) ? NC : Nrt;

    __shared__ float part[8][16];

    const int tid  = threadIdx.x;
    const int wave = tid >> 5;
    const int lane = tid & 31;
    const int slabLocal = wave >> 2;      // 0..1
    const int chunk     = wave & 3;       // 0..3  (K split)
    const int j0 = blockIdx.x * 32 + slabLocal * 16;
    const int l  = lane & 15;             // column within slab / M row
    const int kb = (lane >> 4) * 2;       // K sub-position per ISA A/B layout
    const bool sel = (l == 0);            // only lanes holding A row 0

    v8f c = {0.f, 0.f, 0.f, 0.f, 0.f, 0.f, 0.f, 0.f};

    const int kcnt  = N >> 2;             // rows per chunk = N/4
    const int iters = kcnt >> 2;          // WMMAs per wave (4 rows each)
    const int i0    = chunk * kcnt;

    // Hoisted streaming pointers: B tile walks 4 rows (= 4*N floats) per WMMA.
    const float* pb = TS + (size_t)(i0 + kb) * N + j0 + l;
    const float* pa = nv_in + i0 + kb;

#pragma unroll 8
    for (int it = 0; it < iters; ++it) {
        // Branch-free A build: broadcast loads (one cacheline wave-wide),
        // then cndmask-select so only A row 0 carries nv, rows 1..15 are 0.
        const float t0 = pa[0];
        const float t1 = pa[1];
        v2f a, b;
        b.x = pb[0];
        b.y = pb[N];
        a.x = sel ? t0 : 0.f;
        a.y = sel ? t1 : 0.f;
        // Warm L0 ahead on the L2-resident TS stream (global_prefetch_b8).
        __builtin_prefetch(pb + 16 * (size_t)N, 0, 3);
        c = __builtin_amdgcn_wmma_f32_16x16x4_f32(
                /*neg_a=*/false, a, /*neg_b=*/false, b,
                /*c_mod=*/(short)0, c, /*reuse_a=*/false, /*reuse_b=*/false);
        pa += 4;
        pb += (size_t)4 * N;
    }

    // Row 0 of C lives in accumulator VGPR 0, lanes 0..15.
    if (lane < 16) part[wave][lane] = c[0];
    __syncthreads();

    if (tid < 32) {
        const int nd = blockIdx.x * 32 + tid;     // node index
        const int sl = tid >> 4;
        const int ln = tid & 15;
        float coup = part[sl * 4 + 0][ln] + part[sl * 4 + 1][ln]
                   + part[sl * 4 + 2][ln] + part[sl * 4 + 3][ln];
        const float d = dg[nd];
        coup -= nv_in[nd] * d;                    // coupling = nv@TS - nv*diagTS

        const float dt = times[t + 1] - times[t];
        float b2 = betas[nd];  b2 *= b2;
        float d2 = deltas[nd]; d2 *= d2;
        float c2 = cs[nd];     c2 *= c2;
        float p2 = ps[nd];     p2 *= p2;

        const float u = Us[nd], w = Is[nd], v = Vs[nd];
        const float bUV = b2 * u * v;
        const float kU = -bUV;
        const float kI = bUV - d2 * w;
        const float kV = p2 * w - c2 * v - d + coup;

        const float wk = (dt / 6.f) * ((stage == 1 || stage == 4) ? 1.f : 2.f);
        float sU, sI, sV;
        if (stage == 1) { sU = wk * kU;            sI = wk * kI;            sV = wk * kV; }
        else            { sU = ksU[nd] + wk * kU;  sI = ksI[nd] + wk * kI;  sV = ksV[nd] + wk * kV; }
        ksU[nd] = sU; ksI[nd] = sI; ksV[nd] = sV;

        if (stage < 4) {
            const float cn = (stage == 3) ? dt : 0.5f * dt;
            const float un = U[nd] + cn * kU;
            const float wn = I[nd] + cn * kI;
            const float vn = V[nd] + cn * kV;
            Us[nd] = un; Is[nd] = wn; Vs[nd] = vn;
            nv_out[nd] = nrm[nd] * vn;
        } else {
            const float un = U[nd] + sU;
            const float wn = I[nd] + sI;
            const float vn = V[nd] + sV;
            U[nd] = un;  I[nd] = wn;  V[nd] = vn;
            Us[nd] = un; Is[nd] = wn; Vs[nd] = vn;
            nv_out[nd] = nrm[nd] * vn;
            const size_t base = (size_t)(t + 1) * 3 * N;
            traj[base + nd]         = un;
            traj[base + N + nd]     = wn;
            traj[base + 2 * N + nd] = vn;
        }
    }
}

// ---------------------------------------------------------------------------
// final_kernel: host_sum[t] = sum over nodes; SIR = (host_sum @ W.T).T / N.
// One block per time step; deterministic tree reduction.
// ---------------------------------------------------------------------------
__global__ void final_kernel(const float* __restrict__ traj,
                             const float* __restrict__ W,
                             float* __restrict__ out, int N, int T) {
    __shared__ float red[256];
    __shared__ float h[3];
    const int t = blockIdx.x;
    for (int comp = 0; comp < 3; ++comp) {
        float s = 0.f;
        const size_t base = ((size_t)t * 3 + comp) * N;
        for (int i = threadIdx.x; i < N; i += blockDim.x) s += traj[base + i];
        red[threadIdx.x] = s;
        __syncthreads();
        for (int off = 128; off > 0; off >>= 1) {
            if (threadIdx.x < off) red[threadIdx.x] += red[threadIdx.x + off];
            __syncthreads();
        }
        if (threadIdx.x == 0) h[comp] = red[0];
        __syncthreads();
    }
    if (threadIdx.x < 3) {
        const int cpt = threadIdx.x;
        const float acc = W[cpt * 3 + 0] * h[0] + W[cpt * 3 + 1] * h[1] + W[cpt * 3 + 2] * h[2];
        out[(size_t)cpt * T + t] = acc / (float)N;
    }
}

// ---------------------------------------------------------------------------
extern "C" void kernel_launch(void* const* d_in, const int* in_sizes, int n_in,
                              void* d_out, int out_size, void* d_ws, size_t ws_size,
                              hipStream_t stream) {
    const float* betas  = (const float*)d_in[0];
    const float* deltas = (const float*)d_in[1];
    const float* cs     = (const float*)d_in[2];
    const float* ps     = (const float*)d_in[3];
    const float* ts     = (const float*)d_in[4];
    const float* x0     = (const float*)d_in[5];
    const float* W      = (const float*)d_in[6];
    const float* times  = (const float*)d_in[7];
    float* out = (float*)d_out;

    const int N = in_sizes[0];
    const int T = in_sizes[7];

    // Workspace layout (floats): TS(N*N) | nrm | dg | U I V | Us Is Vs |
    //                            nvA nvB | ksU ksI ksV | traj(T*3N)
    float* w   = (float*)d_ws;
    float* TS  = w;                 w += (size_t)N * N;
    float* nrm = w;                 w += N;
    float* dg  = w;                 w += N;
    float* U   = w;                 w += N;
    float* I   = w;                 w += N;
    float* V   = w;                 w += N;
    float* Us  = w;                 w += N;
    float* Is  = w;                 w += N;
    float* Vs  = w;                 w += N;
    float* nvA = w;                 w += N;
    float* nvB = w;                 w += N;
    float* ksU = w;                 w += N;
    float* ksI = w;                 w += N;
    float* ksV = w;                 w += N;
    float* traj = w;                // T * 3N

    prep_matrix<<<N, 256, 0, stream>>>(ts, TS, nrm, dg, N);
    prep_state<<<(N + 255) / 256, 256, 0, stream>>>(x0, nrm, U, I, V, Us, Is, Vs, nvA, traj, N);

    int parity = 0;  // nvA holds nv for the upcoming stage when parity==0
    for (int t = 0; t < T - 1; ++t) {
        for (int st = 1; st <= 4; ++st) {
            float* nin  = (parity == 0) ? nvA : nvB;
            float* nout = (parity == 0) ? nvB : nvA;
            if (N == 4096) {
                stage_kernel<4096><<<N / 32, 256, 0, stream>>>(
                    TS, nrm, dg, betas, deltas, cs, ps,
                    U, I, V, Us, Is, Vs, nin, nout, ksU, ksI, ksV,
                    traj, times, t, st, N);
            } else {
                stage_kernel<0><<<N / 32, 256, 0, stream>>>(
                    TS, nrm, dg, betas, deltas, cs, ps,
                    U, I, V, Us, Is, Vs, nin, nout, ksU, ksI, ksV,
                    traj, times, t, st, N);
            }
            parity ^= 1;
        }
    }

    final_kernel<<<T, 256, 0, stream>>>(traj, W, out, N, T);
}